// GCN_10282151706722
// MI455X (gfx1250) — compile-verified
//
#include <hip/hip_runtime.h>
#include <hip/hip_bf16.h>

typedef __attribute__((ext_vector_type(2))) float v2f;
typedef __attribute__((ext_vector_type(8))) float v8f;

// ---------------------------------------------------------------------------
// Zero the SpMM accumulators (AH, AH1) in workspace. float4 stores.
// ---------------------------------------------------------------------------
__global__ __launch_bounds__(256) void zero4_kernel(float4* __restrict__ p, int n4) {
    int i = blockIdx.x * 256 + threadIdx.x;
    if (i < n4) p[i] = float4{0.f, 0.f, 0.f, 0.f};
}

// ---------------------------------------------------------------------------
// Edge-parallel SpMM: AH[row[e],:] += vals[e] * H[col[e],:]
// D/4 lanes per edge, float4 gather + 4x global_atomic_add_f32 scatter.
// H / AH are L2-resident on MI455X (192MB L2), so this runs at L2 rates.
// ---------------------------------------------------------------------------
template <int D>
__global__ __launch_bounds__(256) void spmm_atomic_kernel(
    const int* __restrict__ row, const int* __restrict__ col,
    const float* __restrict__ vals, const float* __restrict__ Hsrc,
    float* __restrict__ AHdst, int nEdges)
{
    constexpr int TPE = D / 4;                 // threads per edge (16 or 32)
    unsigned int tid = blockIdx.x * 256u + threadIdx.x;
    int e = (int)(tid / TPE);
    int c = (int)(tid % TPE);
    if (e >= nEdges) return;

    // stream the index/value arrays through the prefetcher (global_prefetch_b8)
    if (c == 0 && (e + 4096) < nEdges) {
        __builtin_prefetch(&row[e + 4096], 0, 0);
        __builtin_prefetch(&col[e + 4096], 0, 0);
        __builtin_prefetch(&vals[e + 4096], 0, 0);
    }

    int r = row[e];
    int s = col[e];
    float v = vals[e];

    const float4 h = *reinterpret_cast<const float4*>(Hsrc + (size_t)s * D + c * 4);
    float* dst = AHdst + (size_t)r * D + c * 4;
    atomicAdd(dst + 0, v * h.x);
    atomicAdd(dst + 1, v * h.y);
    atomicAdd(dst + 2, v * h.z);
    atomicAdd(dst + 3, v * h.w);
}

// ---------------------------------------------------------------------------
// Fused GEMM + bias + ReLU with V_WMMA_F32_16X16X4_F32.
// Out[M x 128] = relu(A[M x K] @ W[K x 128] + b), M multiple of 16.
// Block = 256 threads = 8 waves. Block owns a 16-row slab of A (staged in
// LDS); wave w computes the 16x16 tile at columns [16w, 16w+16).
//
// f32 WMMA operand layout (ISA 7.12.2):
//   A 16x4 : lanes 0-15 -> row M=lane,    {K=0,K=1}; lanes 16-31 -> {K=2,K=3}
//   B 4x16 : lanes 0-15 -> col N=lane,    {K=0,K=1}; lanes 16-31 -> {K=2,K=3}
//   C/D    : VGPR i, lanes 0-15 -> M=i, N=lane; lanes 16-31 -> M=8+i, N=lane-16
// ---------------------------------------------------------------------------
template <int K>
__global__ __launch_bounds__(256) void gemm_bias_relu_kernel(
    const float* __restrict__ A, const float* __restrict__ W,
    const float* __restrict__ bias, float* __restrict__ out)
{
    __shared__ float As[16 * K];

    const int tid = threadIdx.x;
    const int blockRow = blockIdx.x;           // which 16-row slab

    // Cooperatively stage the 16 x K A-slab into LDS (float4 per load).
    {
        const float4* Ag = reinterpret_cast<const float4*>(A + (size_t)blockRow * 16 * K);
        float4* As4 = reinterpret_cast<float4*>(As);
        constexpr int NV4 = 16 * K / 4;        // 256 (K=64) or 512 (K=128)
        #pragma unroll
        for (int i = tid; i < NV4; i += 256) As4[i] = Ag[i];
    }
    __syncthreads();

    const int wave = tid >> 5;                 // 0..7 -> N tile index
    const int lane = tid & 31;
    const int half = lane >> 4;                // 0: K=0,1 ; 1: K=2,3
    const int l15  = lane & 15;
    const int n    = wave * 16 + l15;          // output column (0..127)

    v8f acc = {};
    const float* Wcol = W + n;                 // column n of W, row stride 128

    #pragma unroll
    for (int k = 0; k < K; k += 4) {
        const int kk = k + 2 * half;
        // A fragment: two consecutive K values of row l15 (8B-aligned ds_load)
        v2f a = *reinterpret_cast<const v2f*>(&As[l15 * K + kk]);
        // B fragment: two consecutive K rows of column n
        v2f b;
        b.x = Wcol[(size_t)kk * 128];
        b.y = Wcol[(size_t)(kk + 1) * 128];
        acc = __builtin_amdgcn_wmma_f32_16x16x4_f32(
            /*neg_a=*/false, a, /*neg_b=*/false, b,
            /*c_mod=*/(short)0, acc, /*reuse_a=*/false, /*reuse_b=*/false);
    }

    const float bv = bias[n];
    const int rowBase = blockRow * 16 + half * 8;
    #pragma unroll
    for (int i = 0; i < 8; ++i) {
        float v = acc[i] + bv;
        out[(size_t)(rowBase + i) * 128 + n] = v > 0.f ? v : 0.f;
    }
}

// ---------------------------------------------------------------------------
// Launch: zero ws accumulators, SpMM1, GEMM1, SpMM2, GEMM2.
// d_ws layout: AH [N*64 f32] | AH1 [N*128 f32] | H1 [N*128 f32]  (128 MB)
// ---------------------------------------------------------------------------
extern "C" void kernel_launch(void* const* d_in, const int* in_sizes, int n_in,
                              void* d_out, int out_size, void* d_ws, size_t ws_size,
                              hipStream_t stream) {
    const int*   row  = (const int*)d_in[0];
    const int*   col  = (const int*)d_in[1];
    const float* vals = (const float*)d_in[2];
    const float* H    = (const float*)d_in[3];
    const float* W1   = (const float*)d_in[4];
    const float* b1   = (const float*)d_in[5];
    const float* W2   = (const float*)d_in[6];
    const float* b2   = (const float*)d_in[7];
    float* out = (float*)d_out;

    const int E = in_sizes[0];            // 1,600,000
    const int N = in_sizes[3] / 64;       // 100,000 (multiple of 16)

    float* AH  = (float*)d_ws;
    float* AH1 = AH  + (size_t)N * 64;
    float* H1  = AH1 + (size_t)N * 128;

    // Zero AH and AH1 (contiguous: N*64 + N*128 floats).
    {
        int n4 = (N * 64 + N * 128) / 4;  // 4.8M float4
        zero4_kernel<<<(n4 + 255) / 256, 256, 0, stream>>>((float4*)d_ws, n4);
    }

    // Layer 0: AH = A @ H  (D=64, 16 lanes/edge)
    {
        long long threads = (long long)E * 16;
        int blocks = (int)((threads + 255) / 256);
        spmm_atomic_kernel<64><<<blocks, 256, 0, stream>>>(row, col, vals, H, AH, E);
    }
    // H1 = relu(AH @ W1 + b1)
    gemm_bias_relu_kernel<64><<<N / 16, 256, 0, stream>>>(AH, W1, b1, H1);

    // Layer 1: AH1 = A @ H1  (D=128, 32 lanes/edge)
    {
        long long threads = (long long)E * 32;
        int blocks = (int)((threads + 255) / 256);
        spmm_atomic_kernel<128><<<blocks, 256, 0, stream>>>(row, col, vals, H1, AH1, E);
    }
    // H2 = relu(AH1 @ W2 + b2) -> d_out
    gemm_bias_relu_kernel<128><<<N / 16, 256, 0, stream>>>(AH1, W2, b2, out);
}